// ByteShiftPowerOf2_7945689497934
// MI455X (gfx1250) — compile-verified
//
#include <hip/hip_runtime.h>
#include <cstdint>
#include <cstddef>

#define D_FEAT          128
#define WAVES_PER_BLOCK 8
#define ROWS_PER_WAVE   8
#define ROWS_PER_BLOCK  (WAVES_PER_BLOCK * ROWS_PER_WAVE)

// Raw 32-bit LDS byte offset (addrspace(3) pointer value) for async-LDS ops.
__device__ __forceinline__ unsigned lds_off_u32(const void* p) {
    return (unsigned)(size_t)(__attribute__((address_space(3))) const void*)p;
}

// CDNA5 async global->LDS copy, GVS mode: saddr = wave-uniform row base,
// vaddr = per-lane byte offset, vdst = per-lane LDS byte address.
__device__ __forceinline__ void async_row_load(unsigned lds_byte_addr,
                                               unsigned lane_byte_off,
                                               const float* row_base) {
    asm volatile("global_load_async_to_lds_b128 %0, %1, %2 offset:0"
                 :
                 : "v"(lds_byte_addr), "v"(lane_byte_off),
                   "s"((unsigned long long)(uintptr_t)row_base)
                 : "memory");
}

__device__ __forceinline__ int bperm_i(int src_lane, int v) {
    return __builtin_amdgcn_ds_bpermute(src_lane << 2, v);
}
__device__ __forceinline__ float bperm_f(int src_lane, float v) {
    return __int_as_float(
        __builtin_amdgcn_ds_bpermute(src_lane << 2, __float_as_int(v)));
}

__global__ __launch_bounds__(256)
void ByteShiftPowerOf2_kernel(const float* __restrict__ in,
                              float* __restrict__ out, int nrows) {
    __shared__ float4 lds[WAVES_PER_BLOCK][2][32];   // 8 KB: 2x512B per wave

    const int lane = threadIdx.x & 31;
    const int wave = threadIdx.x >> 5;
    const long long rowBase =
        ((long long)blockIdx.x * WAVES_PER_BLOCK + wave) * ROWS_PER_WAVE;

    const unsigned voff = (unsigned)(lane * 16);
    const unsigned ldsA0 = lds_off_u32(&lds[wave][0][lane]);
    const unsigned ldsA1 = lds_off_u32(&lds[wave][1][lane]);

    if (rowBase < nrows)
        async_row_load(ldsA0, voff, in + (size_t)rowBase * D_FEAT);

#pragma unroll
    for (int i = 0; i < ROWS_PER_WAVE; ++i) {
        const long long row = rowBase + i;
        const bool valid  = row < nrows;
        const bool nvalid = (i + 1 < ROWS_PER_WAVE) && (row + 1 < nrows);

        if (nvalid)
            async_row_load((((i + 1) & 1) ? ldsA1 : ldsA0), voff,
                           in + (size_t)(row + 1) * D_FEAT);

        if (nvalid) asm volatile("s_wait_asynccnt 0x1" ::: "memory");
        else        asm volatile("s_wait_asynccnt 0x0" ::: "memory");

        if (!valid) return;   // wave-uniform; later rows also invalid

        // Lane l holds features 4l..4l+3 of this row.
        float4 f = lds[wave][i & 1][lane];

        // First-occurrence argmax over this lane's 4 elements.
        float bv = f.x; int bi = lane * 4;
        if (f.y > bv) { bv = f.y; bi = lane * 4 + 1; }
        if (f.z > bv) { bv = f.z; bi = lane * 4 + 2; }
        if (f.w > bv) { bv = f.w; bi = lane * 4 + 3; }

        // XOR butterfly within aligned groups of 4 lanes (= 16 features).
        // Tie-break: strictly greater wins, equal value -> lower index.
#pragma unroll
        for (int m = 1; m <= 2; m <<= 1) {
            int   ol = lane ^ m;
            float ov = bperm_f(ol, bv);
            int   oi = bperm_i(ol, bi);
            if (ov > bv || (ov == bv && oi < bi)) { bv = ov; bi = oi; }
        }
        // lanes 4-7: argmax over [16,32); 8-11: [32,48); 12-15: [48,64)
        int val_lo = __builtin_amdgcn_readlane(bi, 4)  - 16;
        int val_hi = __builtin_amdgcn_readlane(bi, 8)  - 32;
        int shift  = __builtin_amdgcn_readlane(bi, 12) - 48;
        if (shift > 31) shift = 31;

        float x0 = __int_as_float(__builtin_amdgcn_readlane(__float_as_int(f.x), 0));
        float x1 = __int_as_float(__builtin_amdgcn_readlane(__float_as_int(f.y), 0));
        float x2 = __int_as_float(__builtin_amdgcn_readlane(__float_as_int(f.z), 0));

        bool shl    = x1 > 0.5f;
        bool active = (x0 >= 0.5f) && (shl || (x2 > 0.5f));

        int value  = val_lo | (val_hi << 4);                       // 0..255
        int result = shl ? ((value << shift) & 255) : (value >> shift);
        int pos_lo = 64 + (result & 15);                           // in [64,79]
        int pos_hi = 80 + (result >> 4);                           // in [80,95]

        float4 o = f;
        if (active) {                                              // uniform branch
            int b0 = lane * 4;
            o.x += ((b0     == pos_lo) || (b0     == pos_hi)) ? 2.0f : 0.0f;
            o.y += ((b0 + 1 == pos_lo) || (b0 + 1 == pos_hi)) ? 2.0f : 0.0f;
            o.z += ((b0 + 2 == pos_lo) || (b0 + 2 == pos_hi)) ? 2.0f : 0.0f;
            o.w += ((b0 + 3 == pos_lo) || (b0 + 3 == pos_hi)) ? 2.0f : 0.0f;
        }
        ((float4*)out)[(size_t)row * 32 + lane] = o;               // coalesced 512B
    }
}

extern "C" void kernel_launch(void* const* d_in, const int* in_sizes, int n_in,
                              void* d_out, int out_size, void* d_ws, size_t ws_size,
                              hipStream_t stream) {
    (void)n_in; (void)out_size; (void)d_ws; (void)ws_size;
    const float* x = (const float*)d_in[0];
    float* out     = (float*)d_out;
    const int n      = in_sizes[0];
    const int nrows  = n / D_FEAT;
    const int blocks = (nrows + ROWS_PER_BLOCK - 1) / ROWS_PER_BLOCK;
    ByteShiftPowerOf2_kernel<<<blocks, 256, 0, stream>>>(x, out, nrows);
}